// GNNAttention_13709535608836
// MI455X (gfx1250) — compile-verified
//
#include <hip/hip_runtime.h>
#include <hip/hip_bf16.h>
#include <math.h>

// Problem constants (match reference)
#define B_    2048
#define NN_   1000
#define L_    20
#define LB_   30
#define ES_   34
#define SES_  12
#define NF_   2
// NFEAT = 34+2+12+1+1 = 50; base (n-independent) part = 48

typedef __attribute__((ext_vector_type(2))) float v2f;
typedef __attribute__((ext_vector_type(8))) float v8f;

__global__ __launch_bounds__(256)
void gnn_fused_kernel(const int*   __restrict__ stops,      // (B,L)
                      const float* __restrict__ x,          // (B,LB,NN)
                      const float* __restrict__ x_dist,     // (NN)
                      const float* __restrict__ x_features, // (B,NF)
                      const int*   __restrict__ x_week,     // (B)
                      const unsigned char* __restrict__ x_mask, // (B,NN) bool
                      const float* __restrict__ stop_emb,   // (NN,SES)
                      const float* __restrict__ week_emb,   // (NW,ES)
                      const float* __restrict__ fc1_w,      // (1,LB)
                      const float* __restrict__ fc1_b,      // (1)
                      const float* __restrict__ lin_l_w,    // (1,50)
                      const float* __restrict__ lin_r_w,    // (1,50)
                      const float* __restrict__ lin_b,      // (1)
                      float*       __restrict__ out)        // (B,NN)
{
    __shared__ float s_val[NN_];        // out[b,n], then logits in-place
    __shared__ float s_w[32];           // fc1 weights padded K=32
    __shared__ int   s_edst[L_ - 1];    // edge dst nodes
    __shared__ float s_eout[L_ - 1];    // out[b, src_e]
    __shared__ float s_edist[L_ - 1];   // x_dist[src_e]
    __shared__ float s_red[8];          // per-wave reduction scratch
    __shared__ float s_scal[4];         // [0]=dot_l [1]=dot_r [2]=max [3]=sumexp

    const int b    = blockIdx.x;
    const int tid  = threadIdx.x;
    const int lane = tid & 31;
    const int wave = tid >> 5;

    // ---- phase 0: init ----
    if (tid < 32) s_w[tid] = (tid < LB_) ? fc1_w[tid] : 0.0f;
    if (tid < 4)  s_scal[tid] = 0.0f;
    __syncthreads();

    // ---- phase A: n-independent feature dot products (48 lanes) ----
    // feat layout: [0..33]=week_emb row, [34..35]=x_features, [36..47]=x_embed
    if (tid < 48) {
        float v;
        if (tid < ES_) {
            v = week_emb[(size_t)x_week[b] * ES_ + tid];
        } else if (tid < ES_ + NF_) {
            v = x_features[(size_t)b * NF_ + (tid - ES_)];
        } else {
            const int j = tid - (ES_ + NF_);          // 0..11
            float s = 0.0f;
            #pragma unroll
            for (int e = 0; e < L_; ++e)
                s += stop_emb[(size_t)stops[b * L_ + e] * SES_ + j];
            v = s;
        }
        atomicAdd(&s_scal[0], v * lin_l_w[tid]);      // ds_add_f32
        atomicAdd(&s_scal[1], v * lin_r_w[tid]);
    }

    // ---- phase B: out[b,n] = sum_l x[b,l,n]*w[l] + b0 via V_WMMA_F32_16X16X4_F32
    // A (16x4): only row M=0 carries weights -> lanes 0 (K=k0,k0+1) and 16 (K=k0+2,k0+3).
    //           Loop-invariant: hoisted into registers ONCE so the hot loop has no
    //           LDS traffic or exec-mask churn.
    // B (4x16): lane&15 = column n; half-wave selects K pair -> two coalesced 64B
    //           row segments of x per load step (NT: streamed once, don't rinse L2).
    // D (16x16): row M=0 => VGPR0 of lanes 0..15 holds out[n0..n0+15].
    {
        const float  bias  = fc1_b[0];
        const int    half2 = (lane >> 4) * 2;         // 0 or 2
        const bool   arow  = ((lane & 15) == 0);      // lanes 0 and 16 carry A row M=0
        const int    ncol  = (lane & 15);
        const size_t xbase = (size_t)b * (LB_ * NN_);
        const int    ntile = (NN_ + 15) / 16;         // 63

        // Preload loop-invariant A fragments (weights) into registers.
        v2f am[8];
        #pragma unroll
        for (int kb = 0; kb < 8; ++kb) {
            const int ka = kb * 4 + half2;            // s_w padded to 32 with zeros
            am[kb].x = arow ? s_w[ka]     : 0.0f;
            am[kb].y = arow ? s_w[ka + 1] : 0.0f;
        }

        for (int tile = wave; tile < ntile; tile += 8) {
            const int n0 = tile * 16;
            const int nc = n0 + ncol;
            const int nclamp = (nc < NN_) ? nc : (NN_ - 1);   // keep loads in-bounds
            const float* xp = x + xbase + nclamp;

            v8f acc = {};
            #pragma unroll
            for (int kb = 0; kb < 8; ++kb) {
                const int ka = kb * 4 + half2;        // K index of .x component
                v2f bm;
                bm.x = (ka     < LB_) ? __builtin_nontemporal_load(xp + (size_t)ka * NN_)       : 0.0f;
                bm.y = (ka + 1 < LB_) ? __builtin_nontemporal_load(xp + (size_t)(ka + 1) * NN_) : 0.0f;
                acc = __builtin_amdgcn_wmma_f32_16x16x4_f32(
                          /*neg_a=*/false, am[kb], /*neg_b=*/false, bm,
                          /*c_mod=*/(short)0, acc,
                          /*reuse_a=*/false, /*reuse_b=*/false);
            }
            if ((lane >> 4) == 0 && nc < NN_)
                s_val[nc] = acc[0] + bias;            // D[0, n]
        }
    }
    __syncthreads();

    // ---- phase C: 19-edge gather table ----
    if (tid < L_ - 1) {
        const int s = stops[b * L_ + tid];
        const int d = stops[b * L_ + tid + 1];
        s_edst[tid]  = d;
        s_eout[tid]  = s_val[s];
        s_edist[tid] = x_dist[s];
    }
    __syncthreads();

    // ---- phase D: per-node logits + block max ----
    const float dot_l = s_scal[0];
    const float dot_r = s_scal[1];
    const float wl48 = lin_l_w[48], wl49 = lin_l_w[49];
    const float wr48 = lin_r_w[48], wr49 = lin_r_w[49];
    const float lb0  = lin_b[0];

    float lmax = -INFINITY;
    for (int n = tid; n < NN_; n += 256) {
        const float o = s_val[n];
        float cnt = 0.0f, ao = 0.0f, ad = 0.0f;
        #pragma unroll
        for (int e = 0; e < L_ - 1; ++e) {
            if (s_edst[e] == n) { cnt += 1.0f; ao += s_eout[e]; ad += s_edist[e]; }
        }
        const float gl = (cnt > 0.0f) ? (dot_l + (ao * wl48 + ad * wl49) / cnt) : 0.0f;
        const float logit = lb0 + dot_r + o * wr48 + x_dist[n] * wr49 + gl;
        s_val[n] = logit;
        lmax = fmaxf(lmax, logit);
    }
    #pragma unroll
    for (int off = 16; off > 0; off >>= 1)
        lmax = fmaxf(lmax, __shfl_xor(lmax, off, 32));
    if (lane == 0) s_red[wave] = lmax;
    __syncthreads();
    if (tid == 0) {
        float m = s_red[0];
        #pragma unroll
        for (int i = 1; i < 8; ++i) m = fmaxf(m, s_red[i]);
        s_scal[2] = m;
    }
    __syncthreads();

    // ---- phase E: sum of exp, log-softmax, masked store ----
    const float gmax = s_scal[2];
    float lsum = 0.0f;
    for (int n = tid; n < NN_; n += 256)
        lsum += __expf(s_val[n] - gmax);
    #pragma unroll
    for (int off = 16; off > 0; off >>= 1)
        lsum += __shfl_xor(lsum, off, 32);
    if (lane == 0) atomicAdd(&s_scal[3], lsum);
    __syncthreads();

    const float  logz = gmax + __logf(s_scal[3]);
    const size_t ob   = (size_t)b * NN_;
    for (int n = tid; n < NN_; n += 256) {
        const float lp = s_val[n] - logz;
        out[ob + n] = x_mask[ob + n] ? -100000000.0f : lp;
    }
}

extern "C" void kernel_launch(void* const* d_in, const int* in_sizes, int n_in,
                              void* d_out, int out_size, void* d_ws, size_t ws_size,
                              hipStream_t stream)
{
    (void)in_sizes; (void)n_in; (void)out_size; (void)d_ws; (void)ws_size;

    const int*   stops      = (const int*)  d_in[0];
    const float* x          = (const float*)d_in[1];
    const float* x_dist     = (const float*)d_in[2];
    const float* x_features = (const float*)d_in[3];
    const int*   x_week     = (const int*)  d_in[4];
    const unsigned char* x_mask = (const unsigned char*)d_in[5]; // jax bool -> 1 byte
    const float* stop_emb   = (const float*)d_in[6];
    const float* week_emb   = (const float*)d_in[7];
    const float* fc1_w      = (const float*)d_in[8];
    const float* fc1_b      = (const float*)d_in[9];
    const float* lin_l_w    = (const float*)d_in[10];
    const float* lin_r_w    = (const float*)d_in[11];
    const float* lin_b      = (const float*)d_in[12];
    float*       out        = (float*)d_out;

    gnn_fused_kernel<<<B_, 256, 0, stream>>>(
        stops, x, x_dist, x_features, x_week, x_mask,
        stop_emb, week_emb, fc1_w, fc1_b, lin_l_w, lin_r_w, lin_b, out);
}